// MemoryCortex_25262997636009
// MI455X (gfx1250) — compile-verified
//
#include <hip/hip_runtime.h>
#include <math.h>

#define T_  16
#define B_  8
#define S_  256
#define D_  1024
#define M_  1024
#define H_  8
#define HD_ 128
#define N_  (B_*S_)          // 2048

typedef __attribute__((ext_vector_type(2))) float     v2f;
typedef __attribute__((ext_vector_type(8))) float     v8f;
typedef __attribute__((ext_vector_type(4))) unsigned  v4u;
typedef __attribute__((ext_vector_type(8))) int       v8i;
typedef __attribute__((ext_vector_type(4))) int       v4i;

__device__ __forceinline__ v8f wmma_f32_4(v2f a, v2f b, v8f c) {
  // V_WMMA_F32_16X16X4_F32: (neg_a, A, neg_b, B, c_mod, C, reuse_a, reuse_b)
  return __builtin_amdgcn_wmma_f32_16x16x4_f32(false, a, false, b, (short)0, c,
                                               false, false);
}

// --------------------------------------------------------- TDM tile loader
// Loads a 64x64 fp32 tile (row stride 1024 floats in memory) into LDS with a
// padded row stride of 68 floats (pad_interval=5 -> every 64 DWORDs,
// pad_amount=3 -> 4 DWORDs), via the Tensor Data Mover.  One call == one D#.
__device__ __forceinline__ void tdm_load_tile_64x64(const float* gptr,
                                                    float* lds_base,
                                                    int rows_total) {
  size_t ga = (size_t)gptr;
  v4u g0;
  g0.x = 1u;                                            // count=1, user mode
  g0.y = (unsigned)(size_t)lds_base;                    // LDS byte address
  g0.z = (unsigned)ga;                                  // global_addr[31:0]
  g0.w = (unsigned)((ga >> 32) & 0x01FFFFFFull)         // global_addr[56:32]
         | (2u << 30);                                  // type=2 ("image")
  v8i g1;
  g1[0] = (int)((2u << 16)      // data_size = 4 bytes
                | (1u << 20)    // pad_enable
                | (5u << 22)    // pad_interval: 2^5 * 8B = 64 DWORDs
                | (3u << 25));  // pad_amount: 4 DWORDs
  g1[1] = (int)(1024u << 16);                     // tensor_dim0=1024 (lo16<<16)
  g1[2] = (int)(((unsigned)rows_total & 0xFFFFu) << 16);   // dim0_hi=0|dim1_lo
  g1[3] = (int)((((unsigned)rows_total >> 16) & 0xFFFFu)   // dim1_hi
                | (64u << 16));                            // tile_dim0=64
  g1[4] = 64;                                     // tile_dim1=64, tile_dim2=0
  g1[5] = 1024;                                   // tensor_dim0_stride[31:0]
  g1[6] = 0;                                      // stride0_hi | stride1_lo
  g1[7] = 0;                                      // stride1_hi
  v4i z4 = {};
  v8i z8 = {};
  __builtin_amdgcn_tensor_load_to_lds(g0, g1, z4, z4, z8, 0);
}

// ---------------------------------------------------------------- utilities
__global__ void zero_f32_kernel(float* __restrict__ p, int n) {
  int i = blockIdx.x * blockDim.x + threadIdx.x;
  if (i < n) p[i] = 0.0f;
}

// Qflat[m] = sum_hd read_query[h(m),hd] * W_read_key[h(m)*HD+hd, m]
__global__ void qflat_kernel(const float* __restrict__ Wrk,
                             const float* __restrict__ rq,
                             float* __restrict__ qflat) {
  int m = blockIdx.x * blockDim.x + threadIdx.x;
  if (m >= M_) return;
  int h = m / HD_;
  float acc = 0.0f;
  for (int hd = 0; hd < HD_; ++hd)
    acc += rq[h * HD_ + hd] * Wrk[(size_t)(h * HD_ + hd) * M_ + m];
  qflat[m] = acc;
}

// ------------------------------------------- fused write-path GEMMs + LIF
// grid = (M/64, N/64), block = 256 (8 waves; wave -> 16n x 32m accumulators)
__global__ __launch_bounds__(256)
void fused_write_lif_kernel(const float* __restrict__ x,       // (T,N,D)
                            const float* __restrict__ Wm,      // (M,D)
                            const float* __restrict__ Wg,      // (M,D)
                            const float* __restrict__ qflat,   // (M)
                            const float* __restrict__ gate_thr,
                            float* __restrict__ mv_out,        // (T,N,M)
                            float* __restrict__ gate_out,      // (N,M)
                            float* __restrict__ logits) {      // (T,N,H)
  __shared__ float xs[64 * 68];
  __shared__ float wsm[64 * 68];
  __shared__ float wsg[64 * 68];

  const int m0   = blockIdx.x * 64;
  const int n0   = blockIdx.y * 64;
  const int tid  = threadIdx.x;
  const int wave = tid >> 5;
  const int lane = tid & 31;
  const int wn   = wave & 3;      // n sub-tile (16 rows each)
  const int wm   = wave >> 2;     // m sub-tile (32 cols each)
  const int lh   = lane & 15;
  const int hiq  = lane >> 4;     // lane-half selects K pair / row group

  const int mg0 = m0 + wm * 32 + lh;        // sub-tile 0 column
  const int mg1 = mg0 + 16;                 // sub-tile 1 column
  const int headq = (m0 + wm * 32) / HD_;   // 32-wide chunk never crosses head
  const float qv0 = qflat[mg0];
  const float qv1 = qflat[mg1];

  v8f vmem0 = {}, vmem1 = {}, vgate0 = {}, vgate1 = {}, gacc0 = {}, gacc1 = {};

  for (int t = 0; t < T_; ++t) {
    v8f mi0 = {}, mi1 = {}, gi0 = {}, gi1 = {};
    const float* xt = x + (size_t)t * N_ * D_;

    for (int d0 = 0; d0 < D_; d0 += 64) {
      __syncthreads();                       // LDS free (prev compute done)
      if (tid < 32) {                        // wave 0 drives the TDM
        tdm_load_tile_64x64(&xt[(size_t)n0 * D_ + d0], xs,  N_);
        tdm_load_tile_64x64(&Wm[(size_t)m0 * D_ + d0], wsm, M_);
        tdm_load_tile_64x64(&Wg[(size_t)m0 * D_ + d0], wsg, M_);
        __builtin_amdgcn_s_wait_tensorcnt(0);
      }
      __syncthreads();                       // tiles visible to all waves

      const int arow  = (wn * 16 + lh) * 68;
      const int brow0 = (wm * 32 + lh) * 68;
      const int brow1 = (wm * 32 + 16 + lh) * 68;
#pragma unroll
      for (int k0 = 0; k0 < 64; k0 += 4) {
        int ks = k0 + hiq * 2;               // lane half holds K pair {0,1}/{2,3}
        v2f a   = *(const v2f*)&xs[arow + ks];
        v2f bm0 = *(const v2f*)&wsm[brow0 + ks];
        v2f bm1 = *(const v2f*)&wsm[brow1 + ks];
        v2f bg0 = *(const v2f*)&wsg[brow0 + ks];
        v2f bg1 = *(const v2f*)&wsg[brow1 + ks];
        mi0 = wmma_f32_4(a, bm0, mi0);
        mi1 = wmma_f32_4(a, bm1, mi1);
        gi0 = wmma_f32_4(a, bg0, gi0);
        gi1 = wmma_f32_4(a, bg1, gi1);
      }
    }

    // ---- LIF update (fp32, hard threshold at VTH=1) + mv trace store ----
    v8f mvv0, mvv1;
#pragma unroll
    for (int j = 0; j < 8; ++j) {
      float v0 = 0.99f * vmem0[j] + mi0[j];
      float s0 = (v0 >= 1.0f) ? 1.0f : 0.0f;
      mvv0[j] = v0;  vmem0[j] = v0 - s0;
      float v1 = 0.99f * vmem1[j] + mi1[j];
      float s1 = (v1 >= 1.0f) ? 1.0f : 0.0f;
      mvv1[j] = v1;  vmem1[j] = v1 - s1;

      float g0 = 0.9f * vgate0[j] + gi0[j];
      float sg0 = (g0 >= 1.0f) ? 1.0f : 0.0f;
      gacc0[j] += sg0;  vgate0[j] = g0 - sg0;
      float g1 = 0.9f * vgate1[j] + gi1[j];
      float sg1 = (g1 >= 1.0f) ? 1.0f : 0.0f;
      gacc1[j] += sg1;  vgate1[j] = g1 - sg1;
    }

    float* mvt = mv_out + (size_t)t * N_ * M_;
    const int nrow = n0 + wn * 16 + hiq * 8;
#pragma unroll
    for (int j = 0; j < 8; ++j) {
      mvt[(size_t)(nrow + j) * M_ + mg0] = mvv0[j];
      mvt[(size_t)(nrow + j) * M_ + mg1] = mvv1[j];
    }

    // ---- attention logit partials: reduce over this wave's 32 m-cols ----
    float c[8];
#pragma unroll
    for (int j = 0; j < 8; ++j) c[j] = mvv0[j] * qv0 + mvv1[j] * qv1;
#pragma unroll
    for (int mask = 1; mask < 16; mask <<= 1)
#pragma unroll
      for (int j = 0; j < 8; ++j) c[j] += __shfl_xor(c[j], mask, 32);

    float* lg = logits + (size_t)t * N_ * H_;
#pragma unroll
    for (int j = 0; j < 8; ++j)
      if (lh == j)
        atomicAdd(&lg[(size_t)(nrow + j) * H_ + headq], c[j]);
  }

  // ---- gate mask = sigmoid((mean_t spikes - thr) * 10) ----
  const float thr = gate_thr[0];
  const int nrow = n0 + wn * 16 + hiq * 8;
#pragma unroll
  for (int j = 0; j < 8; ++j) {
    float g0 = 1.0f / (1.0f + expf(-10.0f * (gacc0[j] * (1.0f / 16.0f) - thr)));
    float g1 = 1.0f / (1.0f + expf(-10.0f * (gacc1[j] * (1.0f / 16.0f) - thr)));
    gate_out[(size_t)(nrow + j) * M_ + mg0] = g0;
    gate_out[(size_t)(nrow + j) * M_ + mg1] = g1;
  }
}

// ---------------------------------------------------------- softmax over T
__global__ void softmax_t_kernel(const float* __restrict__ logits,
                                 float* __restrict__ attnw) {
  int idx = blockIdx.x * blockDim.x + threadIdx.x;   // n*H + h
  if (idx >= N_ * H_) return;
  const float scale = 0.088388347648318447f;         // 1/sqrt(128)
  float v[T_], mx = -1e30f;
#pragma unroll
  for (int t = 0; t < T_; ++t) {
    v[t] = logits[(size_t)t * N_ * H_ + idx] * scale;
    mx = fmaxf(mx, v[t]);
  }
  float s = 0.0f;
#pragma unroll
  for (int t = 0; t < T_; ++t) { v[t] = expf(v[t] - mx); s += v[t]; }
  float inv = 1.0f / s;
#pragma unroll
  for (int t = 0; t < T_; ++t)
    attnw[(size_t)t * N_ * H_ + idx] = v[t] * inv;
}

// ----------------------------------- gated attention-weighted read over T
__global__ __launch_bounds__(256)
void mem_read_kernel(const float* __restrict__ mv,
                     const float* __restrict__ attnw,
                     const float* __restrict__ gate,
                     float* __restrict__ mread) {
  int idx = blockIdx.x * blockDim.x + threadIdx.x;   // over N*M/4
  if (idx >= N_ * (M_ / 4)) return;
  int n  = idx / (M_ / 4);
  int m4 = (idx - n * (M_ / 4)) * 4;
  int h  = m4 / HD_;
  float4 acc = make_float4(0.f, 0.f, 0.f, 0.f);
#pragma unroll
  for (int t = 0; t < T_; ++t) {
    float w = attnw[(size_t)t * N_ * H_ + (size_t)n * H_ + h];
    float4 m = *(const float4*)&mv[(size_t)t * N_ * M_ + (size_t)n * M_ + m4];
    acc.x += w * m.x; acc.y += w * m.y; acc.z += w * m.z; acc.w += w * m.w;
  }
  float4 g = *(const float4*)&gate[(size_t)n * M_ + m4];
  acc.x *= g.x; acc.y *= g.y; acc.z *= g.z; acc.w *= g.w;
  *(float4*)&mread[(size_t)n * M_ + m4] = acc;
}

// --------------------------------------------- output GEMM (N,M)x(D,M)^T
// grid = (D/64, N/64), block = 256
__global__ __launch_bounds__(256)
void out_gemm_kernel(const float* __restrict__ mread,  // (N,M)
                     const float* __restrict__ Wf,     // (D,M)
                     float* __restrict__ mout) {       // (N,D)
  __shared__ float as_[64 * 68];
  __shared__ float bs_[64 * 68];
  const int d0c = blockIdx.x * 64;
  const int n0  = blockIdx.y * 64;
  const int tid = threadIdx.x;
  const int wave = tid >> 5, lane = tid & 31;
  const int wn = wave & 3, wd = wave >> 2;
  const int lh = lane & 15, hiq = lane >> 4;

  v8f c0 = {}, c1 = {};
  for (int k0 = 0; k0 < M_; k0 += 64) {
    __syncthreads();
    if (tid < 32) {
      tdm_load_tile_64x64(&mread[(size_t)n0 * M_ + k0], as_, N_);
      tdm_load_tile_64x64(&Wf[(size_t)d0c * M_ + k0],   bs_, D_);
      __builtin_amdgcn_s_wait_tensorcnt(0);
    }
    __syncthreads();

    const int arow  = (wn * 16 + lh) * 68;
    const int brow0 = (wd * 32 + lh) * 68;
    const int brow1 = (wd * 32 + 16 + lh) * 68;
#pragma unroll
    for (int k = 0; k < 64; k += 4) {
      int ks = k + hiq * 2;
      v2f a  = *(const v2f*)&as_[arow + ks];
      v2f b0 = *(const v2f*)&bs_[brow0 + ks];
      v2f b1 = *(const v2f*)&bs_[brow1 + ks];
      c0 = wmma_f32_4(a, b0, c0);
      c1 = wmma_f32_4(a, b1, c1);
    }
  }
  const int nrow = n0 + wn * 16 + hiq * 8;
  const int dg0 = d0c + wd * 32 + lh;
#pragma unroll
  for (int j = 0; j < 8; ++j) {
    mout[(size_t)(nrow + j) * D_ + dg0]      = c0[j];
    mout[(size_t)(nrow + j) * D_ + dg0 + 16] = c1[j];
  }
}

// ------------------------------------ LayerNorm + residual broadcast (T)
// grid = N, block = 256 (each thread owns 4 d's)
__global__ __launch_bounds__(256)
void ln_residual_kernel(const float* __restrict__ x,
                        const float* __restrict__ mout,
                        const float* __restrict__ gamma,
                        const float* __restrict__ beta,
                        const float* __restrict__ mixp,
                        float* __restrict__ out) {
  __shared__ float r1[8], r2[8];
  const int n = blockIdx.x, tid = threadIdx.x;
  const int wave = tid >> 5, lane = tid & 31;
  const int d4 = tid * 4;

  float4 v = *(const float4*)&mout[(size_t)n * D_ + d4];
  float s1 = v.x + v.y + v.z + v.w;
  float s2 = v.x * v.x + v.y * v.y + v.z * v.z + v.w * v.w;
#pragma unroll
  for (int off = 16; off > 0; off >>= 1) {
    s1 += __shfl_xor(s1, off, 32);
    s2 += __shfl_xor(s2, off, 32);
  }
  if (lane == 0) { r1[wave] = s1; r2[wave] = s2; }
  __syncthreads();
  if (wave == 0) {
    float a = (lane < 8) ? r1[lane] : 0.0f;
    float b = (lane < 8) ? r2[lane] : 0.0f;
#pragma unroll
    for (int off = 4; off > 0; off >>= 1) {
      a += __shfl_xor(a, off, 32);
      b += __shfl_xor(b, off, 32);
    }
    if (lane == 0) { r1[0] = a; r2[0] = b; }
  }
  __syncthreads();
  const float mu  = r1[0] * (1.0f / D_);
  const float var = r2[0] * (1.0f / D_) - mu * mu;
  const float rs  = rsqrtf(var + 1e-5f);
  const float mix = 1.0f / (1.0f + expf(-mixp[0]));

  float4 gm = *(const float4*)&gamma[d4];
  float4 bt = *(const float4*)&beta[d4];
  float4 nv;
  nv.x = ((v.x - mu) * rs * gm.x + bt.x) * mix;
  nv.y = ((v.y - mu) * rs * gm.y + bt.y) * mix;
  nv.z = ((v.z - mu) * rs * gm.z + bt.z) * mix;
  nv.w = ((v.w - mu) * rs * gm.w + bt.w) * mix;

#pragma unroll
  for (int t = 0; t < T_; ++t) {
    size_t off = (size_t)t * N_ * D_ + (size_t)n * D_ + d4;
    float4 xv = *(const float4*)&x[off];
    xv.x += nv.x; xv.y += nv.y; xv.z += nv.z; xv.w += nv.w;
    *(float4*)&out[off] = xv;
  }
}

// ---------------------------------------------------------------- launcher
extern "C" void kernel_launch(void* const* d_in, const int* in_sizes, int n_in,
                              void* d_out, int out_size, void* d_ws,
                              size_t ws_size, hipStream_t stream) {
  const float* x    = (const float*)d_in[0];
  const float* Wm   = (const float*)d_in[1];
  const float* Wg   = (const float*)d_in[2];
  const float* Wrk  = (const float*)d_in[3];
  const float* Wf   = (const float*)d_in[4];
  const float* rq   = (const float*)d_in[5];
  const float* thr  = (const float*)d_in[6];
  const float* gam  = (const float*)d_in[7];
  const float* bet  = (const float*)d_in[8];
  const float* mix  = (const float*)d_in[9];
  float* out = (float*)d_out;

  float* ws = (float*)d_ws;
  const size_t SZ_MV = (size_t)T_ * N_ * M_;   // 33.5M
  const size_t SZ_NM = (size_t)N_ * M_;        //  2.1M
  const size_t SZ_AT = (size_t)T_ * N_ * H_;   //  256K
  float* mv     = ws;
  float* gate   = mv + SZ_MV;
  float* logits = gate + SZ_NM;
  float* attnw  = logits + SZ_AT;
  float* qflat  = attnw + SZ_AT;
  float* mread  = qflat + M_;
  float* mout   = mread + SZ_NM;

  zero_f32_kernel<<<(int)((SZ_AT + 255) / 256), 256, 0, stream>>>(logits,
                                                                 (int)SZ_AT);
  qflat_kernel<<<(M_ + 255) / 256, 256, 0, stream>>>(Wrk, rq, qflat);

  fused_write_lif_kernel<<<dim3(M_ / 64, N_ / 64), 256, 0, stream>>>(
      x, Wm, Wg, qflat, thr, mv, gate, logits);

  softmax_t_kernel<<<(N_ * H_ + 255) / 256, 256, 0, stream>>>(logits, attnw);

  mem_read_kernel<<<(N_ * (M_ / 4) + 255) / 256, 256, 0, stream>>>(mv, attnw,
                                                                   gate, mread);

  out_gemm_kernel<<<dim3(D_ / 64, N_ / 64), 256, 0, stream>>>(mread, Wf, mout);

  ln_residual_kernel<<<N_, 256, 0, stream>>>(x, mout, gam, bet, mix, out);
}